// GCN_15865609191547
// MI455X (gfx1250) — compile-verified
//
#include <hip/hip_runtime.h>

#define NNODES 50000
#define NEDGES 800000
#define NGRAPHS 256

typedef __attribute__((ext_vector_type(2))) float v2f;
typedef __attribute__((ext_vector_type(8))) float v8f;

static inline int cdiv(long long a, int b) { return (int)((a + (long long)b - 1) / b); }

// ---------------- elementwise helpers ----------------
__global__ __launch_bounds__(256) void k_zero(float* __restrict__ p, int n) {
  int i = blockIdx.x * 256 + threadIdx.x;
  if (i < n) p[i] = 0.0f;
}

__global__ __launch_bounds__(256) void k_deg(const int* __restrict__ col,
                                             const float* __restrict__ w,
                                             float* __restrict__ deg, int E) {
  int i = blockIdx.x * 256 + threadIdx.x;
  if (i < E) atomicAdd(&deg[col[i]], w[i]);
}

__global__ __launch_bounds__(256) void k_dinv(float* __restrict__ deg, int n) {
  int i = blockIdx.x * 256 + threadIdx.x;
  if (i < n) {
    float d = deg[i] + 1.0f;               // self-loop weight 1
    deg[i] = (d > 0.0f) ? rsqrtf(d) : 0.0f;
  }
}

__global__ __launch_bounds__(256) void k_init_bias(float* __restrict__ out,
                                                   const float* __restrict__ bias,
                                                   int total, int cmask) {
  int i = blockIdx.x * 256 + threadIdx.x;
  if (i < total) out[i] = bias[i & cmask];
}

__global__ __launch_bounds__(256) void k_add_bias(float* __restrict__ x,
                                                  const float* __restrict__ b,
                                                  int total, int cmask, int relu) {
  int i = blockIdx.x * 256 + threadIdx.x;
  if (i < total) {
    float v = x[i] + b[i & cmask];
    x[i] = (relu && v < 0.0f) ? 0.0f : v;
  }
}

// pad rows: xp[N x dpad] = x[N x din] zero-extended in the channel dim
__global__ __launch_bounds__(256)
void k_pad_rows(const float* __restrict__ x, float* __restrict__ xp,
                int total, int din, int dpad) {
  int i = blockIdx.x * 256 + threadIdx.x;
  if (i >= total) return;
  int r = i / dpad;
  int c = i - r * dpad;
  xp[i] = (c < din) ? x[(size_t)r * din + c] : 0.0f;
}

// pad weight rows: wp[dpad x N] = w[din x N] zero-extended (row-major -> prefix copy)
__global__ __launch_bounds__(256)
void k_pad_prefix(const float* __restrict__ w, float* __restrict__ wp,
                  int total, int valid) {
  int i = blockIdx.x * 256 + threadIdx.x;
  if (i < total) wp[i] = (i < valid) ? w[i] : 0.0f;
}

// ---------------- fp32 WMMA GEMM: C[M,N] = A[M,K] @ B[K,N] ----------------
// One wave32 per 16x16 tile. Compile-time K (multiple of 4) => fully unrolled,
// unconditional loads, dual accumulators to break the WMMA RAW chain.
template <int K>
__global__ __launch_bounds__(256)
void gemm_wmma(const float* __restrict__ A, const float* __restrict__ B,
               float* __restrict__ C, int M, int N) {
  int wave = (blockIdx.x * 256 + threadIdx.x) >> 5;   // wave32
  int lane = threadIdx.x & 31;
  int tilesN = N >> 4;
  int tiles  = (M >> 4) * tilesN;
  if (wave >= tiles) return;                          // wave-uniform exit
  int tm = wave / tilesN;
  int tn = wave - tm * tilesN;

  int khalf = (lane >> 4) << 1;                       // 0 or 2
  const float* Arow = A + (size_t)((tm << 4) + (lane & 15)) * K + khalf;
  const float* Bcol = B + (tn << 4) + (lane & 15) + (size_t)khalf * N;

  v8f c0 = {0.f, 0.f, 0.f, 0.f, 0.f, 0.f, 0.f, 0.f};
  v8f c1 = {0.f, 0.f, 0.f, 0.f, 0.f, 0.f, 0.f, 0.f};
#pragma unroll
  for (int k0 = 0; k0 < K; k0 += 8) {
    v2f a0 = *(const v2f*)(Arow + k0);                // 8B-aligned (K, khalf even)
    v2f b0;
    b0.x = Bcol[(size_t)k0 * N];
    b0.y = Bcol[(size_t)(k0 + 1) * N];
    c0 = __builtin_amdgcn_wmma_f32_16x16x4_f32(false, a0, false, b0, (short)0,
                                               c0, false, false);
    if (k0 + 4 < K) {                                 // compile-time resolved
      v2f a1 = *(const v2f*)(Arow + k0 + 4);
      v2f b1;
      b1.x = Bcol[(size_t)(k0 + 4) * N];
      b1.y = Bcol[(size_t)(k0 + 5) * N];
      c1 = __builtin_amdgcn_wmma_f32_16x16x4_f32(false, a1, false, b1, (short)0,
                                                 c1, false, false);
    }
  }
  int rbase = (tm << 4) + ((lane >> 4) << 3);
  int ncol  = (tn << 4) + (lane & 15);
#pragma unroll
  for (int v = 0; v < 8; ++v)
    C[(size_t)(rbase + v) * N + ncol] = c0[v] + c1[v];
}

// runtime-K fallback (K must be a multiple of 4; guaranteed by padding)
__global__ __launch_bounds__(256)
void gemm_wmma_rt(const float* __restrict__ A, const float* __restrict__ B,
                  float* __restrict__ C, int M, int K, int N) {
  int wave = (blockIdx.x * 256 + threadIdx.x) >> 5;
  int lane = threadIdx.x & 31;
  int tilesN = N >> 4;
  int tiles  = (M >> 4) * tilesN;
  if (wave >= tiles) return;
  int tm = wave / tilesN;
  int tn = wave - tm * tilesN;
  int khalf = (lane >> 4) << 1;
  const float* Arow = A + (size_t)((tm << 4) + (lane & 15)) * K + khalf;
  const float* Bcol = B + (tn << 4) + (lane & 15) + (size_t)khalf * N;
  v8f c = {0.f, 0.f, 0.f, 0.f, 0.f, 0.f, 0.f, 0.f};
  for (int k0 = 0; k0 < K; k0 += 4) {
    v2f a = *(const v2f*)(Arow + k0);
    v2f b;
    b.x = Bcol[(size_t)k0 * N];
    b.y = Bcol[(size_t)(k0 + 1) * N];
    c = __builtin_amdgcn_wmma_f32_16x16x4_f32(false, a, false, b, (short)0, c,
                                              false, false);
  }
  int rbase = (tm << 4) + ((lane >> 4) << 3);
  int ncol  = (tn << 4) + (lane & 15);
#pragma unroll
  for (int v = 0; v < 8; ++v)
    C[(size_t)(rbase + v) * N + ncol] = c[v];
}

static void launch_gemm(const float* A, const float* B, float* C,
                        int M, int K, int N, hipStream_t s) {
  int blocks = cdiv((long long)(M / 16) * (N / 16) * 32, 256);
  switch (K) {
    case 116: gemm_wmma<116><<<blocks, 256, 0, s>>>(A, B, C, M, N); break;
    case 104: gemm_wmma<104><<<blocks, 256, 0, s>>>(A, B, C, M, N); break;
    case 128: gemm_wmma<128><<<blocks, 256, 0, s>>>(A, B, C, M, N); break;
    case 64:  gemm_wmma<64> <<<blocks, 256, 0, s>>>(A, B, C, M, N); break;
    case 32:  gemm_wmma<32> <<<blocks, 256, 0, s>>>(A, B, C, M, N); break;
    default:  gemm_wmma_rt  <<<blocks, 256, 0, s>>>(A, B, C, M, K, N); break;
  }
}

// ---------------- message gather + atomic scatter ----------------
__global__ __launch_bounds__(256)
void k_aggregate(const float* __restrict__ xw,
                 const int* __restrict__ row, const int* __restrict__ col,
                 const float* __restrict__ ew, const float* __restrict__ dinv,
                 float* __restrict__ agg, int E, int Nn, int dout) {
  int chunks = dout >> 2;
  long long tid = (long long)blockIdx.x * 256 + threadIdx.x;
  long long total = (long long)(E + Nn) * chunks;
  if (tid >= total) return;
  int e  = (int)(tid / chunks);
  int c4 = (int)(tid - (long long)e * chunks) << 2;
  int r, cdst; float norm;
  if (e < E) {
    r = row[e]; cdst = col[e];
    norm = dinv[r] * ew[e] * dinv[cdst];
  } else {
    r = e - E; cdst = r;
    float d = dinv[r]; norm = d * d;                 // self loop, weight 1
  }
  const float4 v = *(const float4*)(xw + (size_t)r * dout + c4);
  float* dst = agg + (size_t)cdst * dout + c4;
  atomicAdd(dst + 0, v.x * norm);
  atomicAdd(dst + 1, v.y * norm);
  atomicAdd(dst + 2, v.z * norm);
  atomicAdd(dst + 3, v.w * norm);
}

// ---------------- BatchNorm (training-mode, biased var) ----------------
__global__ __launch_bounds__(256)
void k_bn_stats(const float* __restrict__ h, float* __restrict__ stats,
                int Nn, int dout) {
  __shared__ float s_sum[256], s_sq[256];
  int t = threadIdx.x;
  int c = t & (dout - 1);
  int rows_per = 256 / dout;
  float sum = 0.0f, sq = 0.0f;
  int r = blockIdx.x * rows_per + (t / dout);
  int rstride = gridDim.x * rows_per;
  for (; r < Nn; r += rstride) {
    float v = h[(size_t)r * dout + c];
    sum += v; sq += v * v;
  }
  s_sum[t] = sum; s_sq[t] = sq;
  __syncthreads();
  if (t < dout) {
    float a = 0.0f, b = 0.0f;
    for (int i = t; i < 256; i += dout) { a += s_sum[i]; b += s_sq[i]; }
    atomicAdd(&stats[c], a);
    atomicAdd(&stats[dout + c], b);
  }
}

__global__ void k_bn_finalize(float* __restrict__ stats,
                              const float* __restrict__ gamma,
                              const float* __restrict__ beta, int Nn, int dout) {
  int c = threadIdx.x;
  if (c >= dout) return;
  float mean = stats[c] / (float)Nn;
  float var  = stats[dout + c] / (float)Nn - mean * mean;
  float inv  = rsqrtf(var + 1e-5f);
  float sc   = gamma[c] * inv;
  stats[c]        = sc;                    // scale
  stats[dout + c] = beta[c] - mean * sc;   // shift
}

__global__ __launch_bounds__(256)
void k_bn_apply(float* __restrict__ h, const float* __restrict__ ss,
                int total, int cmask, int dout) {
  int i = blockIdx.x * 256 + threadIdx.x;
  if (i < total) {
    int c = i & cmask;
    float v = h[i] * ss[c] + ss[dout + c];
    h[i] = (v > 0.0f) ? v : 0.0f;
  }
}

// ---------------- segment-mean pooling ----------------
__global__ __launch_bounds__(256)
void k_pool_accum(const float* __restrict__ h, const int* __restrict__ batch,
                  float* __restrict__ sums, float* __restrict__ cnt, int n) {
  int tid = blockIdx.x * 256 + threadIdx.x;
  if (tid >= n * 4) return;
  int node = tid >> 2;
  int c8 = (tid & 3) << 3;
  int g = batch[node];
  const float* src = h + (size_t)node * 32 + c8;
  float* dst = sums + g * 32 + c8;
#pragma unroll
  for (int j = 0; j < 8; ++j) atomicAdd(dst + j, src[j]);
  if ((tid & 3) == 0) atomicAdd(&cnt[g], 1.0f);
}

__global__ __launch_bounds__(256)
void k_pool_finish(const float* __restrict__ sums, const float* __restrict__ cnt,
                   float* __restrict__ pooled) {
  int i = blockIdx.x * 256 + threadIdx.x;
  if (i < NGRAPHS * 32) pooled[i] = sums[i] / fmaxf(cnt[i >> 5], 1.0f);
}

// ---------------- host-side branch driver ----------------
struct Bufs {
  float *buf0, *buf1, *buf2, *dinv, *stats, *psum, *pcnt, *ptmp, *xpad, *wpad;
};

static void run_branch(const float* x, int din,
                       const int* row, const int* col, const float* ew,
                       const int* batch, const float* const* P,
                       float* out_pool, float* out_z, float* out_h,
                       const Bufs& B, hipStream_t s) {
  const int N = NNODES, E = NEDGES;
  // degree -> dinv (in-place)
  k_zero<<<cdiv(N, 256), 256, 0, s>>>(B.dinv, N);
  k_deg<<<cdiv(E, 256), 256, 0, s>>>(col, ew, B.dinv, E);
  k_dinv<<<cdiv(N, 256), 256, 0, s>>>(B.dinv, N);

  // if layer-1 K is not a multiple of 4, zero-pad x and W1 once
  const float* x1 = x;
  const float* W1 = P[0];
  int k1 = din;
  if (din % 4 != 0) {
    int dpad = (din + 7) & ~7;
    k_pad_rows<<<cdiv((long long)N * dpad, 256), 256, 0, s>>>(x, B.xpad,
                                                              N * dpad, din, dpad);
    k_pad_prefix<<<cdiv(dpad * 128, 256), 256, 0, s>>>(P[0], B.wpad,
                                                       dpad * 128, din * 128);
    x1 = B.xpad; W1 = B.wpad; k1 = dpad;
  }

  struct Layer { const float* X; const float* W; int din; int dout;
                 float* xw; float* agg; };
  Layer L[3] = {
    { x1,     W1,   k1,  128, B.buf0, B.buf1 },
    { B.buf1, P[4], 128,  64, B.buf0, B.buf2 },
    { B.buf2, P[8],  64,  32, B.buf0, out_h  },  // layer-3 straight into d_out
  };
  for (int l = 0; l < 3; ++l) {
    const float* bias  = P[l * 4 + 1];
    const float* gamma = P[l * 4 + 2];
    const float* beta  = P[l * 4 + 3];
    int dout = L[l].dout;
    launch_gemm(L[l].X, L[l].W, L[l].xw, N, L[l].din, dout, s);
    int total = N * dout;
    k_init_bias<<<cdiv(total, 256), 256, 0, s>>>(L[l].agg, bias, total, dout - 1);
    long long ath = (long long)(E + N) * (dout / 4);
    k_aggregate<<<cdiv(ath, 256), 256, 0, s>>>(L[l].xw, row, col, ew, B.dinv,
                                               L[l].agg, E, N, dout);
    k_zero<<<1, 256, 0, s>>>(B.stats, 2 * dout);
    k_bn_stats<<<512, 256, 0, s>>>(L[l].agg, B.stats, N, dout);
    k_bn_finalize<<<1, dout, 0, s>>>(B.stats, gamma, beta, N, dout);
    k_bn_apply<<<cdiv(total, 256), 256, 0, s>>>(L[l].agg, B.stats, total,
                                                dout - 1, dout);
  }
  // mean pool (psum and pcnt contiguous: one zero covers both)
  k_zero<<<cdiv(NGRAPHS * 32 + NGRAPHS, 256), 256, 0, s>>>(B.psum,
                                                           NGRAPHS * 32 + NGRAPHS);
  k_pool_accum<<<cdiv((long long)N * 4, 256), 256, 0, s>>>(out_h, batch, B.psum,
                                                           B.pcnt, N);
  k_pool_finish<<<cdiv(NGRAPHS * 32, 256), 256, 0, s>>>(B.psum, B.pcnt, out_pool);
  // projector: relu(pooled @ pW1 + pb1) @ pW2 + pb2 (also via WMMA)
  launch_gemm(out_pool, P[12], B.ptmp, NGRAPHS, 32, 64, s);
  k_add_bias<<<cdiv(NGRAPHS * 64, 256), 256, 0, s>>>(B.ptmp, P[13],
                                                     NGRAPHS * 64, 63, 1);
  launch_gemm(B.ptmp, P[14], out_z, NGRAPHS, 64, 32, s);
  k_add_bias<<<cdiv(NGRAPHS * 32, 256), 256, 0, s>>>(out_z, P[15],
                                                     NGRAPHS * 32, 31, 0);
}

static void fill_params_concat(const float* base, int din, const float** P) {
  int dims[3][2] = { { din, 128 }, { 128, 64 }, { 64, 32 } };
  size_t off = 0; int k = 0;
  for (int l = 0; l < 3; ++l) {
    int di = dims[l][0], dw = dims[l][1];
    P[k++] = base + off; off += (size_t)di * dw;  // W
    P[k++] = base + off; off += dw;               // b
    P[k++] = base + off; off += dw;               // gamma
    P[k++] = base + off; off += dw;               // beta
  }
  P[k++] = base + off; off += 32 * 64;            // pW1
  P[k++] = base + off; off += 64;                 // pb1
  P[k++] = base + off; off += 64 * 32;            // pW2
  P[k++] = base + off; off += 32;                 // pb2
}

extern "C" void kernel_launch(void* const* d_in, const int* in_sizes, int n_in,
                              void* d_out, int out_size, void* d_ws, size_t ws_size,
                              hipStream_t stream) {
  const int N = NNODES, E = NEDGES;
  const float* x_t = (const float*)d_in[0];
  const int*   ei_t = (const int*)d_in[1];
  const float* ea_t = (const float*)d_in[2];
  const int*   bt_t = (const int*)d_in[3];
  const float* x_f = (const float*)d_in[4];
  const int*   ei_f = (const int*)d_in[5];
  const float* ea_f = (const float*)d_in[6];
  const int*   bt_f = (const int*)d_in[7];
  int din_t = in_sizes[0] / N;   // 116
  int din_f = in_sizes[4] / N;   // 101

  const float* Pt[16]; const float* Pf[16];
  if (n_in >= 40) {              // param tuples flattened into separate leaves
    for (int i = 0; i < 16; ++i) Pt[i] = (const float*)d_in[8 + i];
    for (int i = 0; i < 16; ++i) Pf[i] = (const float*)d_in[24 + i];
  } else {                       // param tuples concatenated flat
    fill_params_concat((const float*)d_in[8], din_t, Pt);
    fill_params_concat((const float*)d_in[9], din_f, Pf);
  }

  // workspace carve-up (~86 MB)
  Bufs B;
  float* w = (float*)d_ws;
  B.buf0 = w; w += (size_t)N * 128;
  B.buf1 = w; w += (size_t)N * 128;
  B.buf2 = w; w += (size_t)N * 64;
  B.dinv = w; w += N;
  B.stats = w; w += 256;
  B.psum = w; w += NGRAPHS * 32;
  B.pcnt = w; w += NGRAPHS;
  B.ptmp = w; w += NGRAPHS * 64;
  B.xpad = w; w += (size_t)N * 104;
  B.wpad = w; w += 104 * 128;

  // output carve-up: h_time, z_time, h_freq, z_freq, xt, xf
  float* out = (float*)d_out;
  float* h_time = out;
  float* z_time = out + 8192;
  float* h_freq = out + 16384;
  float* z_freq = out + 24576;
  float* xt     = out + 32768;
  float* xf     = xt + (size_t)N * 32;

  run_branch(x_t, din_t, ei_t, ei_t + E, ea_t, bt_t, Pt,
             h_time, z_time, xt, B, stream);
  run_branch(x_f, din_f, ei_f, ei_f + E, ea_f, bt_f, Pf,
             h_freq, z_freq, xf, B, stream);
}